// WormNetCalcium_79139067396321
// MI455X (gfx1250) — compile-verified
//
#include <hip/hip_runtime.h>
#include <hip/hip_bf16.h>

#define N_NEUR 300
#define WIN    1024
#define T_FULL 16384      // WIN * 16
#define ODOR   16
#define DT_C   0.2f

typedef float v2f __attribute__((ext_vector_type(2)));
typedef float v8f __attribute__((ext_vector_type(8)));

// Async global->LDS copy of 16 bytes per active lane (gfx1250, ASYNCcnt).
__device__ __forceinline__ void async_copy_b128(const void* gptr, void* lptr) {
    unsigned int lds_off = (unsigned int)(unsigned long long)lptr;   // LDS aperture: low 32 bits
    unsigned long long ga = (unsigned long long)gptr;
    asm volatile("global_load_async_to_lds_b128 %0, %1, off"
                 :: "v"(lds_off), "v"(ga) : "memory");
}
__device__ __forceinline__ void async_wait0() {
    asm volatile("s_wait_asynccnt 0x0" ::: "memory");
}

// ---------------------------------------------------------------------------
// Kernel 1: sensory_input[n,t] = (odor[t,:] @ W_enc[n,:] + b_enc[n]) * mask[n]
// ---------------------------------------------------------------------------
__global__ __launch_bounds__(256)
void sensory_kernel(const float* __restrict__ odor,   // (T,16)
                    const float* __restrict__ Wenc,   // (300,16)
                    const float* __restrict__ benc,   // (300)
                    const float* __restrict__ mask,   // (300)
                    float* __restrict__ outS)         // (300,T)
{
    __shared__ float sOdor[256 * ODOR];     // 16 KB
    __shared__ float sW[N_NEUR * ODOR];     // 18.75 KB
    __shared__ float sB[N_NEUR];
    __shared__ float sM[N_NEUR];

    const int tid = threadIdx.x;
    const int t0  = blockIdx.x * 256;

    for (int i = tid; i < 256 * ODOR / 4; i += 256)
        async_copy_b128((const float4*)(odor + (size_t)t0 * ODOR) + i, (float4*)sOdor + i);
    for (int i = tid; i < N_NEUR * ODOR / 4; i += 256)
        async_copy_b128((const float4*)Wenc + i, (float4*)sW + i);
    for (int i = tid; i < N_NEUR; i += 256) { sB[i] = benc[i]; sM[i] = mask[i]; }
    async_wait0();
    __syncthreads();

    const int t = t0 + tid;
    const float4* o = (const float4*)&sOdor[tid * ODOR];
    const float4 o0 = o[0], o1 = o[1], o2 = o[2], o3 = o[3];

    for (int n = 0; n < N_NEUR; ++n) {
        const float4* w = (const float4*)&sW[n * ODOR];
        const float4 w0 = w[0], w1 = w[1], w2 = w[2], w3 = w[3];
        float acc = sB[n];
        acc += o0.x*w0.x + o0.y*w0.y + o0.z*w0.z + o0.w*w0.w;
        acc += o1.x*w1.x + o1.y*w1.y + o1.z*w1.z + o1.w*w1.w;
        acc += o2.x*w2.x + o2.y*w2.y + o2.z*w2.z + o2.w*w2.w;
        acc += o3.x*w3.x + o3.y*w3.y + o3.z*w3.z + o3.w*w3.w;
        outS[(size_t)n * T_FULL + t] = acc * sM[n];
    }
}

// Cooperative async stage of one 16x300 weight block into LDS (75 float4/row).
__device__ __forceinline__ void stage_weight_block(const float* __restrict__ W,
                                                   float* lds, int mRow, int tid) {
    for (int i = tid; i < 16 * 75; i += 256) {
        const int row  = i / 75;
        const int col  = (i - row * 75) * 4;
        int grow = mRow * 16 + row;
        if (grow >= N_NEUR) grow = N_NEUR - 1;     // row-local clamp (stores predicated)
        async_copy_b128(W + (size_t)grow * N_NEUR + col, lds + row * N_NEUR + col);
    }
}

// ---------------------------------------------------------------------------
// Kernel 2 (WMMA): mu_lat = A_mu@f_up + B_mu@S ; lv_lat = A_lv@f_up + B_lv@S
// fused: sample = mu + exp(0.5*lv)*eps ; ca = softplus(aw*sample + ab)
// Weight rows staged once per WG via async global->LDS; A-fragments from LDS
// (ds_load_b64, conflict-free: stride 300 dwords covers all 64 banks).
// ---------------------------------------------------------------------------
__global__ __launch_bounds__(256)
void stageA_kernel(const float* __restrict__ A_mu, const float* __restrict__ B_mu,
                   const float* __restrict__ A_lv, const float* __restrict__ B_lv,
                   const float* __restrict__ raw,   // (300,1024)
                   const float* __restrict__ S,     // (300,T) sensory
                   const float* __restrict__ eps,   // (300,T)
                   const float* __restrict__ aw, const float* __restrict__ ab,
                   float* __restrict__ outMu, float* __restrict__ outLv,
                   float* __restrict__ outSm, float* __restrict__ outCa)
{
    __shared__ float sAmu[16 * N_NEUR];   // 19.2 KB each, 76.8 KB total
    __shared__ float sBmu[16 * N_NEUR];
    __shared__ float sAlv[16 * N_NEUR];
    __shared__ float sBlv[16 * N_NEUR];

    const int tid  = threadIdx.x;
    const int wave = tid >> 5;
    const int lane = tid & 31;
    const int hf   = lane >> 4;
    const int l16  = lane & 15;
    const int mRow = blockIdx.x;
    const int t0   = (blockIdx.y * 8 + wave) * 16;
    const int rawC = t0 >> 4;            // f_up tile == one raw column

    stage_weight_block(A_mu, sAmu, mRow, tid);
    stage_weight_block(B_mu, sBmu, mRow, tid);
    stage_weight_block(A_lv, sAlv, mRow, tid);
    stage_weight_block(B_lv, sBlv, mRow, tid);
    async_wait0();
    __syncthreads();

    const int rowOff = l16 * N_NEUR + 2 * hf;   // A-fragment base (dwords)
    const int koff   = 2 * hf;

    v8f accMu = {0.f,0.f,0.f,0.f,0.f,0.f,0.f,0.f};
    v8f accLv = {0.f,0.f,0.f,0.f,0.f,0.f,0.f,0.f};

    for (int k = 0; k < N_NEUR; k += 4) {
        const v2f aAmu = *(const v2f*)&sAmu[rowOff + k];
        const v2f aBmu = *(const v2f*)&sBmu[rowOff + k];
        const v2f aAlv = *(const v2f*)&sAlv[rowOff + k];
        const v2f aBlv = *(const v2f*)&sBlv[rowOff + k];
        const int k0 = k + koff;
        v2f fb, sb;
        fb.x = raw[(size_t)k0 * WIN + rawC];
        fb.y = raw[(size_t)(k0 + 1) * WIN + rawC];
        sb.x = S[(size_t)k0 * T_FULL + t0 + l16];
        sb.y = S[(size_t)(k0 + 1) * T_FULL + t0 + l16];
        if (k + 12 < N_NEUR)
            __builtin_prefetch(S + (size_t)(k0 + 8) * T_FULL + t0 + l16, 0, 3);
        accMu = __builtin_amdgcn_wmma_f32_16x16x4_f32(false, aAmu, false, fb,
                                                      (short)0, accMu, false, false);
        accMu = __builtin_amdgcn_wmma_f32_16x16x4_f32(false, aBmu, false, sb,
                                                      (short)0, accMu, false, false);
        accLv = __builtin_amdgcn_wmma_f32_16x16x4_f32(false, aAlv, false, fb,
                                                      (short)0, accLv, false, false);
        accLv = __builtin_amdgcn_wmma_f32_16x16x4_f32(false, aBlv, false, sb,
                                                      (short)0, accLv, false, false);
    }

    const int t = t0 + l16;
    #pragma unroll
    for (int r = 0; r < 8; ++r) {
        const int m = mRow * 16 + r + 8 * hf;   // C/D layout: VGPR r, half-wave hf
        if (m < N_NEUR) {
            const size_t idx = (size_t)m * T_FULL + t;
            const float mu = accMu[r];
            const float lv = accLv[r];
            const float sm = mu + expf(0.5f * lv) * eps[idx];
            const float z  = aw[m] * sm + ab[m];
            const float ca = (z > 20.f) ? z : log1pf(expf(z));   // softplus
            outMu[idx] = mu;
            outLv[idx] = lv;
            outSm[idx] = sm;
            outCa[idx] = ca;
        }
    }
}

// ---------------------------------------------------------------------------
// Kernel 3 (WMMA): recurrent_in = W_c@relu(sample) + W_e@sample
// fused: mu_v = sample + alpha*(-sample + rec + S + bias)
// ---------------------------------------------------------------------------
__global__ __launch_bounds__(256)
void stageB_kernel(const float* __restrict__ W_c, const float* __restrict__ W_e,
                   const float* __restrict__ Sm,   // sample (300,T)
                   const float* __restrict__ S,    // sensory (300,T)
                   const float* __restrict__ bias, const float* __restrict__ tau,
                   float* __restrict__ outRec, float* __restrict__ outMuV)
{
    __shared__ float sWc[16 * N_NEUR];    // 38.4 KB total
    __shared__ float sWe[16 * N_NEUR];

    const int tid  = threadIdx.x;
    const int wave = tid >> 5;
    const int lane = tid & 31;
    const int hf   = lane >> 4;
    const int l16  = lane & 15;
    const int mRow = blockIdx.x;
    const int t0   = (blockIdx.y * 8 + wave) * 16;

    stage_weight_block(W_c, sWc, mRow, tid);
    stage_weight_block(W_e, sWe, mRow, tid);
    async_wait0();
    __syncthreads();

    const int rowOff = l16 * N_NEUR + 2 * hf;
    const int koff   = 2 * hf;

    v8f acc = {0.f,0.f,0.f,0.f,0.f,0.f,0.f,0.f};

    for (int k = 0; k < N_NEUR; k += 4) {
        const v2f aWc = *(const v2f*)&sWc[rowOff + k];
        const v2f aWe = *(const v2f*)&sWe[rowOff + k];
        const int k0 = k + koff;
        v2f vb, rb;
        vb.x = Sm[(size_t)k0 * T_FULL + t0 + l16];
        vb.y = Sm[(size_t)(k0 + 1) * T_FULL + t0 + l16];
        rb.x = fmaxf(vb.x, 0.f);
        rb.y = fmaxf(vb.y, 0.f);
        if (k + 12 < N_NEUR)
            __builtin_prefetch(Sm + (size_t)(k0 + 8) * T_FULL + t0 + l16, 0, 3);
        acc = __builtin_amdgcn_wmma_f32_16x16x4_f32(false, aWc, false, rb,
                                                    (short)0, acc, false, false);
        acc = __builtin_amdgcn_wmma_f32_16x16x4_f32(false, aWe, false, vb,
                                                    (short)0, acc, false, false);
    }

    const int t = t0 + l16;
    #pragma unroll
    for (int r = 0; r < 8; ++r) {
        const int m = mRow * 16 + r + 8 * hf;
        if (m < N_NEUR) {
            const size_t idx = (size_t)m * T_FULL + t;
            const float rec   = acc[r];
            const float sm    = Sm[idx];
            const float alpha = DT_C / fmaxf(tau[m], DT_C);
            outRec[idx] = rec;
            outMuV[idx] = sm + alpha * (-sm + rec + S[idx] + bias[m]);
        }
    }
}

// ---------------------------------------------------------------------------
// Kernel 4: exponential calcium filter (first-order linear recurrence).
// One wave per neuron; lane owns a 512-step segment; wave32 shuffle-scan of
// affine transfer (a,b); second pass writes calcium + fluorescence.
// ---------------------------------------------------------------------------
__global__ __launch_bounds__(32)
void calcium_scan_kernel(const float* __restrict__ ca,     // (300,T) activation
                         const float* __restrict__ full,   // (300,T) target
                         const float* __restrict__ scale, const float* __restrict__ shift,
                         const float* __restrict__ ctau,
                         float* __restrict__ outCal, float* __restrict__ outFl)
{
    const int n    = blockIdx.x;
    const int lane = threadIdx.x;
    const int SEG  = T_FULL / 32;

    const float alpha = DT_C / fmaxf(ctau[n], DT_C);
    const float decay = expf(-alpha);
    const float scl   = scale[n];
    const float shf   = shift[n];
    const float init  = (full[(size_t)n * T_FULL] - shf) / scl;

    const size_t base = (size_t)n * T_FULL + (size_t)lane * SEG;

    // pass 1: local segment result from zero state
    float c = 0.f;
    for (int j = 0; j < SEG; ++j) {
        float x = ca[base + j];
        if (lane == 0 && j == 0) x += init;   // c[0] = init + x[0]
        c = decay * c + x;
    }

    // wave32 inclusive scan of affine transfer (a,b): comb(l,r)=(la*ra, lb*ra+rb)
    float a = expf(-alpha * (float)SEG);      // decay^SEG
    float b = c;
    #pragma unroll
    for (int off = 1; off < 32; off <<= 1) {
        const float pa = __shfl_up(a, off, 32);
        const float pb = __shfl_up(b, off, 32);
        if (lane >= off) { b = pb * a + b; a = pa * a; }
    }
    float carry = __shfl_up(b, 1, 32);        // exclusive: state entering segment
    if (lane == 0) carry = 0.f;

    // pass 2: replay segment from carry, emit outputs
    c = carry;
    for (int j = 0; j < SEG; ++j) {
        float x = ca[base + j];
        if (lane == 0 && j == 0) x += init;
        c = decay * c + x;
        outCal[base + j] = c;
        outFl[base + j]  = scl * c + shf;
    }
}

// ---------------------------------------------------------------------------
extern "C" void kernel_launch(void* const* d_in, const int* in_sizes, int n_in,
                              void* d_out, int out_size, void* d_ws, size_t ws_size,
                              hipStream_t stream) {
    const float* raw  = (const float*)d_in[0];   // fluorescence_raw_target (300,1024)
    const float* full = (const float*)d_in[1];   // fluorescence_full_target (300,T)
    const float* odor = (const float*)d_in[2];   // (T,16)
    const float* mask = (const float*)d_in[3];
    const float* Wenc = (const float*)d_in[4];
    const float* benc = (const float*)d_in[5];
    const float* A_mu = (const float*)d_in[6];
    const float* B_mu = (const float*)d_in[7];
    const float* A_lv = (const float*)d_in[8];
    const float* B_lv = (const float*)d_in[9];
    const float* eps  = (const float*)d_in[10];
    const float* W_c  = (const float*)d_in[11];
    const float* W_e  = (const float*)d_in[12];
    const float* bias = (const float*)d_in[13];
    const float* tau  = (const float*)d_in[14];
    const float* aw   = (const float*)d_in[15];
    const float* ab   = (const float*)d_in[16];
    const float* fls  = (const float*)d_in[17];
    const float* flsh = (const float*)d_in[18];
    const float* ctau = (const float*)d_in[19];

    float* out = (float*)d_out;
    const size_t NT = (size_t)N_NEUR * T_FULL;
    float* outMuV = out + 0 * NT;   // mu_v_prob
    float* outFl  = out + 1 * NT;   // mu_fluorescence
    float* outMu  = out + 2 * NT;   // mu_lat
    float* outLv  = out + 3 * NT;   // lv_lat
    float* outSm  = out + 4 * NT;   // sample
    float* outCa  = out + 5 * NT;   // calcium_activation
    float* outCal = out + 6 * NT;   // mu_calcium_prob
    float* outRec = out + 7 * NT;   // recurrent_in
    float* outS   = out + 8 * NT;   // sensory_input

    sensory_kernel<<<T_FULL / 256, 256, 0, stream>>>(odor, Wenc, benc, mask, outS);

    dim3 gemmGrid((N_NEUR + 15) / 16, T_FULL / (16 * 8));   // 19 x 128
    stageA_kernel<<<gemmGrid, 256, 0, stream>>>(A_mu, B_mu, A_lv, B_lv, raw, outS,
                                                eps, aw, ab, outMu, outLv, outSm, outCa);
    stageB_kernel<<<gemmGrid, 256, 0, stream>>>(W_c, W_e, outSm, outS, bias, tau,
                                                outRec, outMuV);
    calcium_scan_kernel<<<N_NEUR, 32, 0, stream>>>(outCa, full, fls, flsh, ctau,
                                                   outCal, outFl);
    (void)in_sizes; (void)n_in; (void)out_size; (void)d_ws; (void)ws_size;
}